// GraphSAGE_81217831568087
// MI455X (gfx1250) — compile-verified
//
#include <hip/hip_runtime.h>

typedef __attribute__((ext_vector_type(2))) float v2f;
typedef __attribute__((ext_vector_type(8))) float v8f;

#define D 128
#define N1C 50000
#define N2C 20000

// o[i] = a[i] + b[i]  (precombine relation weights / biases once per launch)
__global__ __launch_bounds__(256) void vec_add(
    const float* __restrict__ a, const float* __restrict__ b,
    float* __restrict__ o, int n)
{
    int i = blockIdx.x * blockDim.x + threadIdx.x;
    if (i < n) o[i] = a[i] + b[i];
}

// One 32-lane wave per edge: lane handles 4 consecutive features (float4),
// coalesced 512B gather per edge, scatter via global_atomic_add_f32.
__global__ __launch_bounds__(256) void sage_scatter(
    const float* __restrict__ x, const int* __restrict__ src,
    const int* __restrict__ dst, float* __restrict__ hsum,
    float* __restrict__ deg, int E)
{
    long t = (long)blockIdx.x * blockDim.x + threadIdx.x;
    int e = (int)(t >> 5);
    int lane = (int)(t & 31);
    if (e >= E) return;
    int s = src[e];
    int d = dst[e];
    const float4 v = *reinterpret_cast<const float4*>(x + (size_t)s * D + lane * 4);
    float* out = hsum + (size_t)d * D + lane * 4;
    atomicAdd(out + 0, v.x);
    atomicAdd(out + 1, v.y);
    atomicAdd(out + 2, v.z);
    atomicAdd(out + 3, v.w);
    if (lane == 0) atomicAdd(deg + d, 1.0f);
}

// out[M x 128] = relu?( xdst@Ws + (hn0/deg0)@Wn0 + (hn1/deg1)@Wn1 + bias )
// One wave computes a 16-row x 128-col stripe via V_WMMA_F32_16X16X4_F32.
// M must be a multiple of 16 (holds: 50000, 20000). EXEC stays all-ones per wave.
__global__ __launch_bounds__(256) void sage_gemm(
    const float* __restrict__ xdst, const float* __restrict__ Ws,
    const float* __restrict__ hn0, const float* __restrict__ deg0, const float* __restrict__ Wn0,
    const float* __restrict__ hn1, const float* __restrict__ deg1, const float* __restrict__ Wn1,
    const float* __restrict__ bias,
    float* __restrict__ out, int M, int do_relu)
{
    const int lane = threadIdx.x & 31;
    const int wave = threadIdx.x >> 5;
    const int rb   = blockIdx.x * (blockDim.x >> 5) + wave;
    if (rb * 16 >= M) return;                  // whole wave exits together
    const int m0   = rb * 16;
    const int mrow = lane & 15;                // row within tile (A/B/C col index)
    const int kq   = (lane >> 4) << 1;         // K-pair select: 0 or 2

    const float* a0 = xdst + (size_t)(m0 + mrow) * D;
    const float* a1 = hn0  + (size_t)(m0 + mrow) * D;
    const float* a2 = hn1  + (size_t)(m0 + mrow) * D;
    const float r1 = 1.0f / fmaxf(deg0[m0 + mrow], 1.0f);
    const float r2 = 1.0f / fmaxf(deg1[m0 + mrow], 1.0f);

    v8f acc[8];
#pragma unroll
    for (int i = 0; i < 8; ++i) acc[i] = (v8f)0.0f;

    for (int k = 0; k < D; k += 4) {
        v2f fa0, fa1, fa2;
        fa0.x = a0[k + kq];      fa0.y = a0[k + kq + 1];
        fa1.x = a1[k + kq] * r1; fa1.y = a1[k + kq + 1] * r1;
        fa2.x = a2[k + kq] * r2; fa2.y = a2[k + kq + 1] * r2;

        const float* w0 = Ws  + (size_t)(k + kq) * D + mrow;
        const float* w1 = Wn0 + (size_t)(k + kq) * D + mrow;
        const float* w2 = Wn1 + (size_t)(k + kq) * D + mrow;

#pragma unroll
        for (int nt = 0; nt < 8; ++nt) {
            const int nb = nt * 16;
            v2f fb;
            fb.x = w0[nb];  fb.y = w0[D + nb];
            acc[nt] = __builtin_amdgcn_wmma_f32_16x16x4_f32(
                false, fa0, false, fb, (short)0, acc[nt], false, false);
            fb.x = w1[nb];  fb.y = w1[D + nb];
            acc[nt] = __builtin_amdgcn_wmma_f32_16x16x4_f32(
                false, fa1, false, fb, (short)0, acc[nt], false, false);
            fb.x = w2[nb];  fb.y = w2[D + nb];
            acc[nt] = __builtin_amdgcn_wmma_f32_16x16x4_f32(
                false, fa2, false, fb, (short)0, acc[nt], false, false);
        }
    }

    // C/D layout: VGPR i -> rows i (lanes 0-15) and i+8 (lanes 16-31), col = lane&15
    const int rhalf = (lane >> 4) * 8;
#pragma unroll
    for (int nt = 0; nt < 8; ++nt) {
        const int n = nt * 16 + mrow;
        const float bv = bias[n];
#pragma unroll
        for (int i = 0; i < 8; ++i) {
            float v = acc[nt][i] + bv;
            if (do_relu) v = fmaxf(v, 0.0f);
            out[(size_t)(m0 + rhalf + i) * D + n] = v;
        }
    }
}

extern "C" void kernel_launch(void* const* d_in, const int* in_sizes, int n_in,
                              void* d_out, int out_size, void* d_ws, size_t ws_size,
                              hipStream_t stream) {
    (void)n_in; (void)out_size; (void)ws_size;
    const float* x      = (const float*)d_in[0];
    const int*   e0s0   = (const int*)d_in[1];
    const int*   e0d0   = (const int*)d_in[2];
    const int*   e0s1   = (const int*)d_in[3];
    const int*   e0d1   = (const int*)d_in[4];
    const int*   e1s0   = (const int*)d_in[5];
    const int*   e1d0   = (const int*)d_in[6];
    const int*   e1s1   = (const int*)d_in[7];
    const int*   e1d1   = (const int*)d_in[8];
    const float* Ws1_r0 = (const float*)d_in[9];
    const float* Wn1_r0 = (const float*)d_in[10];
    const float* Ws1_r1 = (const float*)d_in[11];
    const float* Wn1_r1 = (const float*)d_in[12];
    const float* Ws2_r0 = (const float*)d_in[13];
    const float* Wn2_r0 = (const float*)d_in[14];
    const float* Ws2_r1 = (const float*)d_in[15];
    const float* Wn2_r1 = (const float*)d_in[16];
    const float* b1_r0  = (const float*)d_in[17];
    const float* b1_r1  = (const float*)d_in[18];
    const float* b2_r0  = (const float*)d_in[19];
    const float* b2_r1  = (const float*)d_in[20];

    const int E0 = in_sizes[1];
    const int E1 = in_sizes[5];

    float* ws = (float*)d_ws;
    size_t o = 0;
    float* hn1_r0  = ws + o; o += (size_t)N1C * D;
    float* hn1_r1  = ws + o; o += (size_t)N1C * D;
    float* deg1_r0 = ws + o; o += N1C;
    float* deg1_r1 = ws + o; o += N1C;
    float* hn2_r0  = ws + o; o += (size_t)N2C * D;
    float* hn2_r1  = ws + o; o += (size_t)N2C * D;
    float* deg2_r0 = ws + o; o += N2C;
    float* deg2_r1 = ws + o; o += N2C;
    const size_t zero_bytes = o * sizeof(float);
    float* h1      = ws + o; o += (size_t)N1C * D;
    float* Wsum1   = ws + o; o += (size_t)D * D;
    float* Wsum2   = ws + o; o += (size_t)D * D;
    float* bsum1   = ws + o; o += D;
    float* bsum2   = ws + o; o += D;

    hipMemsetAsync(d_ws, 0, zero_bytes, stream);

    {   // precombine per-relation self weights and biases (loop-invariant)
        const int nW = D * D;
        vec_add<<<(nW + 255) / 256, 256, 0, stream>>>(Ws1_r0, Ws1_r1, Wsum1, nW);
        vec_add<<<(nW + 255) / 256, 256, 0, stream>>>(Ws2_r0, Ws2_r1, Wsum2, nW);
        vec_add<<<1, 256, 0, stream>>>(b1_r0, b1_r1, bsum1, D);
        vec_add<<<1, 256, 0, stream>>>(b2_r0, b2_r1, bsum2, D);
    }
    {   // layer 1 aggregation (both relations)
        const int threads = 256;
        const long total = (long)E0 * 32;
        const int blocks = (int)((total + threads - 1) / threads);
        sage_scatter<<<blocks, threads, 0, stream>>>(x, e0s0, e0d0, hn1_r0, deg1_r0, E0);
        sage_scatter<<<blocks, threads, 0, stream>>>(x, e0s1, e0d1, hn1_r1, deg1_r1, E0);
    }
    {   // layer 1 GEMM + relu -> h1
        const int rbs = N1C / 16;
        const int blocks = (rbs + 7) / 8;   // 8 waves (16-row stripes) per block
        sage_gemm<<<blocks, 256, 0, stream>>>(x, Wsum1,
            hn1_r0, deg1_r0, Wn1_r0, hn1_r1, deg1_r1, Wn1_r1,
            bsum1, h1, N1C, 1);
    }
    {   // layer 2 aggregation from h1
        const int threads = 256;
        const long total = (long)E1 * 32;
        const int blocks = (int)((total + threads - 1) / threads);
        sage_scatter<<<blocks, threads, 0, stream>>>(h1, e1s0, e1d0, hn2_r0, deg2_r0, E1);
        sage_scatter<<<blocks, threads, 0, stream>>>(h1, e1s1, e1d1, hn2_r1, deg2_r1, E1);
    }
    {   // layer 2 GEMM -> d_out
        const int rbs = N2C / 16;
        const int blocks = (rbs + 7) / 8;
        sage_gemm<<<blocks, 256, 0, stream>>>(h1, Wsum2,
            hn2_r0, deg2_r0, Wn2_r0, hn2_r1, deg2_r1, Wn2_r1,
            bsum2, (float*)d_out, N2C, 0);
    }
}